// SMPL_query_78864189489217
// MI455X (gfx1250) — compile-verified
//
#include <hip/hip_runtime.h>
#include <stdint.h>

#define V_NUM   6890
#define F_NUM   13776
#define NBATCH  2
#define NPTS    1024
#define NPOINTS (NBATCH * NPTS)          // 2048
#define NCHUNK  (F_NUM / 16)             // 861 (13776 = 861*16 exactly)
#define NSTRIP  7
#define CPS     (NCHUNK / NSTRIP)        // 123
#define NGROUP  (NPOINTS / 16)           // 128
#define WPB     4                        // waves per block (128 threads)
#define REC     80                       // float4 per chunk record: 32 B-op + 48 tri

typedef float v2f __attribute__((ext_vector_type(2)));
typedef float v8f __attribute__((ext_vector_type(8)));

// ---------------- workspace layout (bytes) ----------------
// gbest : NPOINTS * u64                                   @ 0       (16384)
// cdata : NBATCH*NCHUNK records of 80 float4              @ 16384   (2204160)
//         record = [32 x {Bx,By,R,R2} WMMA-B lanes | 48 x tri vert float4]
#define WS_CD 16384

__device__ __forceinline__ float safe_div(float n, float d) {
    return n / (fabsf(d) < 1e-12f ? 1.0f : d);
}

// Branchless Ericson region test, identical mask order to the reference.
__device__ __forceinline__ void bary_closest(
    float px, float py, float pz,
    float ax, float ay, float az,
    float bx, float by, float bz,
    float cx, float cy, float cz,
    float& u, float& v, float& w)
{
    float abx = bx - ax, aby = by - ay, abz = bz - az;
    float acx = cx - ax, acy = cy - ay, acz = cz - az;
    float apx = px - ax, apy = py - ay, apz = pz - az;
    float d1 = abx*apx + aby*apy + abz*apz;
    float d2 = acx*apx + acy*apy + acz*apz;
    float bpx = px - bx, bpy = py - by, bpz = pz - bz;
    float d3 = abx*bpx + aby*bpy + abz*bpz;
    float d4 = acx*bpx + acy*bpy + acz*bpz;
    float cpx = px - cx, cpy = py - cy, cpz = pz - cz;
    float d5 = abx*cpx + aby*cpy + abz*cpz;
    float d6 = acx*cpx + acy*cpy + acz*cpz;
    float vc = d1*d4 - d3*d2;
    float vb = d5*d2 - d1*d6;
    float va = d3*d6 - d5*d4;
    float t_ab = safe_div(d1, d1 - d3);
    float t_ac = safe_div(d2, d2 - d6);
    float t_bc = safe_div(d4 - d3, (d4 - d3) + (d5 - d6));
    float denom = va + vb + vc;
    float v_in = safe_div(vb, denom);
    float w_in = safe_div(vc, denom);
    u = 1.0f - v_in - w_in; v = v_in; w = w_in;
    bool m1 = (va <= 0.f) && ((d4 - d3) >= 0.f) && ((d5 - d6) >= 0.f);
    u = m1 ? 0.f : u;          v = m1 ? 1.f - t_bc : v;  w = m1 ? t_bc : w;
    bool m2 = (vb <= 0.f) && (d2 >= 0.f) && (d6 <= 0.f);
    u = m2 ? 1.f - t_ac : u;   v = m2 ? 0.f : v;         w = m2 ? t_ac : w;
    bool m3 = (vc <= 0.f) && (d1 >= 0.f) && (d3 <= 0.f);
    u = m3 ? 1.f - t_ab : u;   v = m3 ? t_ab : v;        w = m3 ? 0.f : w;
    bool m4 = (d6 >= 0.f) && (d5 <= d6);
    u = m4 ? 0.f : u;          v = m4 ? 0.f : v;         w = m4 ? 1.f : w;
    bool m5 = (d3 >= 0.f) && (d4 <= d3);
    u = m5 ? 0.f : u;          v = m5 ? 1.f : v;         w = m5 ? 0.f : w;
    bool m6 = (d1 <= 0.f) && (d2 <= 0.f);
    u = m6 ? 1.f : u;          v = m6 ? 0.f : v;         w = m6 ? 0.f : w;
}

// CDNA5 async copy: global -> LDS of one 80-float4 chunk record, 3 instrs.
// Coverage with 32 lanes: [0..31], [32..63], [48..79] (48..63 duplicated, benign).
__device__ __forceinline__ void stage_issue(float4* lds, const float4* gbase, int lane) {
    uint32_t l0 = (uint32_t)(uintptr_t)(lds + lane);        // low 32 bits of flat LDS addr = LDS offset
    uint32_t l1 = (uint32_t)(uintptr_t)(lds + 32 + lane);
    uint32_t l2 = (uint32_t)(uintptr_t)(lds + 48 + lane);
    const float4* g0 = gbase + lane;
    const float4* g1 = gbase + 32 + lane;
    const float4* g2 = gbase + 48 + lane;
    asm volatile("global_load_async_to_lds_b128 %0, %1, off" :: "v"(l0), "v"(g0) : "memory");
    asm volatile("global_load_async_to_lds_b128 %0, %1, off" :: "v"(l1), "v"(g1) : "memory");
    asm volatile("global_load_async_to_lds_b128 %0, %1, off" :: "v"(l2), "v"(g2) : "memory");
}

// ---------------- kernel 1: per-triangle precompute + best init ----------------
__global__ __launch_bounds__(256) void k_prep(
    const float* __restrict__ smpl_V, const int* __restrict__ smpl_F,
    float4* __restrict__ cdata, unsigned long long* __restrict__ gbest)
{
    int t = blockIdx.x * 256 + threadIdx.x;
    if (t < NPOINTS) gbest[t] = 0x7F7FFFFFFFFFFFFFull;   // (FLT_MAX bits)<<32 | ~0
    if (t >= NBATCH * F_NUM) return;
    int b = t / F_NUM, f = t % F_NUM;
    int ia = smpl_F[f*3+0], ib = smpl_F[f*3+1], ic = smpl_F[f*3+2];
    const float* Vb = smpl_V + (size_t)b * V_NUM * 3;
    float ax = Vb[ia*3+0], ay = Vb[ia*3+1], az = Vb[ia*3+2];
    float bx = Vb[ib*3+0], by = Vb[ib*3+1], bz = Vb[ib*3+2];
    float cx = Vb[ic*3+0], cy = Vb[ic*3+1], cz = Vb[ic*3+2];
    float gx = (ax+bx+cx) * (1.0f/3.0f);
    float gy = (ay+by+cy) * (1.0f/3.0f);
    float gz = (az+bz+cz) * (1.0f/3.0f);
    float ra = (ax-gx)*(ax-gx) + (ay-gy)*(ay-gy) + (az-gz)*(az-gz);
    float rb = (bx-gx)*(bx-gx) + (by-gy)*(by-gy) + (bz-gz)*(bz-gz);
    float rc = (cx-gx)*(cx-gx) + (cy-gy)*(cy-gy) + (cz-gz)*(cz-gz);
    float R2 = fmaxf(ra, fmaxf(rb, rc));
    float R  = sqrtf(R2);
    float q  = gx*gx + gy*gy + gz*gz;
    int chunk = f >> 4, n = f & 15;
    float4* rec = cdata + (size_t)(b * NCHUNK + chunk) * REC;
    rec[n]      = make_float4(-2.f*gx, -2.f*gy, R, R2);   // B rows K=0,1 (lanes 0-15)
    rec[16 + n] = make_float4(-2.f*gz, q,       R, R2);   // B rows K=2,3 (lanes 16-31)
    rec[32 + n*3 + 0] = make_float4(ax, ay, az, 0.f);
    rec[32 + n*3 + 1] = make_float4(bx, by, bz, 0.f);
    rec[32 + n*3 + 2] = make_float4(cx, cy, cz, 0.f);
}

// ---------------- kernel 2: WMMA-pruned brute-force argmin ----------------
__global__ __launch_bounds__(128) void k_search(
    const float* __restrict__ coords, unsigned long long* __restrict__ gbest,
    const float4* __restrict__ cdata)
{
    __shared__ float4 stage[WPB][2][REC];
    __shared__ unsigned long long lbest[WPB][16];
    __shared__ float pps[WPB][16];

    int wv = threadIdx.x >> 5, lane = threadIdx.x & 31;
    int wid = blockIdx.x * WPB + wv;
    int g = wid / NSTRIP, strip = wid % NSTRIP;
    int b = g >> 6;
    int n0 = (g & 63) << 4;
    int half = lane >> 4, m = lane & 15;

    const float* pbase = coords + (size_t)(b * NPTS + n0) * 3;
    float px = pbase[m*3+0], py = pbase[m*3+1], pz = pbase[m*3+2];
    v2f Aop; Aop.x = half ? pz : px; Aop.y = half ? 1.0f : py;   // A 16x4: (px,py,pz,1)

    if (lane < 16) {
        pps[wv][m]   = px*px + py*py + pz*pz;
        lbest[wv][m] = 0x7F7FFFFFFFFFFFFFull;
    }
    asm volatile("s_wait_dscnt 0x0" ::: "memory");
    float ppm[8];
#pragma unroll
    for (int r = 0; r < 8; ++r) ppm[r] = pps[wv][8*half + r];
    float bd2A[8], tbdA[8];
#pragma unroll
    for (int r = 0; r < 8; ++r) { bd2A[r] = 3.4e38f; tbdA[r] = 3.4e38f; }

    int c0 = strip * CPS;
    const float4* crec = cdata + (size_t)b * NCHUNK * REC;
    stage_issue(stage[wv][0], crec + (size_t)c0 * REC, lane);    // prologue prefetch

    for (int i = 0; i < CPS; ++i) {
        int chunk = c0 + i;
        int bi = i & 1;
        asm volatile("s_wait_dscnt 0x0" ::: "memory");   // prior LDS reads done before buffer reuse
        if (i + 1 < CPS) {
            stage_issue(stage[wv][1-bi], crec + (size_t)(chunk + 1) * REC, lane);
            asm volatile("s_wait_asynccnt 0x3" ::: "memory");   // current record resident
        } else {
            asm volatile("s_wait_asynccnt 0x0" ::: "memory");
        }

        const float4* sbuf = stage[wv][bi];
        float4 BR = sbuf[lane];                                 // ds_load_b128, per-lane B-operand
        v2f Bop; Bop.x = BR.x; Bop.y = BR.y;                    // (-2c | q) column
        float Rv = BR.z, Rv2 = BR.w;                            // (R, R^2) of this lane's triangle
        v8f Cz = {0.f,0.f,0.f,0.f,0.f,0.f,0.f,0.f};
        // D[m][n] = ||c_n||^2 - 2 p_m . c_n   (16 points x 16 triangles)
        v8f D = __builtin_amdgcn_wmma_f32_16x16x4_f32(
            false, Aop, false, Bop, (short)0, Cz, false, false);

        // survive iff min_r( ||p_r - c||^2 - (best_r + R)^2 ) <= 0
        float worst = 3.4e38f;
#pragma unroll
        for (int r = 0; r < 8; ++r) {
            float d2pc = D[r] + ppm[r];
            float thr  = fmaf(tbdA[r], Rv, bd2A[r]) + Rv2;
            worst = fminf(worst, d2pc - thr);
        }
        unsigned mask = __builtin_amdgcn_ballot_w32(worst <= 0.f);
        unsigned m16 = (mask | (mask >> 16)) & 0xFFFFu;

        if (m16) {
            const float4* sb = sbuf + 32;
            while (m16) {
                int j1 = __builtin_ctz(m16); m16 &= m16 - 1;
                int j2 = m16 ? __builtin_ctz(m16) : j1;
                m16 = m16 ? (m16 & (m16 - 1)) : 0u;
                int t = half ? j2 : j1;                          // two triangles per pass
                float4 va4 = sb[t*3+0], vb4 = sb[t*3+1], vc4 = sb[t*3+2];
                float u, v, w;
                bary_closest(px, py, pz,
                             va4.x, va4.y, va4.z,
                             vb4.x, vb4.y, vb4.z,
                             vc4.x, vc4.y, vc4.z, u, v, w);
                float hx = u*va4.x + v*vb4.x + w*vc4.x;
                float hy = u*va4.y + v*vb4.y + w*vc4.y;
                float hz = u*va4.z + v*vb4.z + w*vc4.z;
                float dx = px - hx, dy = py - hy, dz = pz - hz;
                float dd = dx*dx + dy*dy + dz*dz;
                unsigned fid = (unsigned)(chunk * 16 + t);
                unsigned long long pk =
                    ((unsigned long long)__float_as_uint(dd) << 32) | (unsigned long long)fid;
                atomicMin(&lbest[wv][m], pk);                    // ds_min_u64, both halves merge
            }
            asm volatile("s_wait_dscnt 0x0" ::: "memory");
#pragma unroll
            for (int r = 0; r < 8; ++r) {
                unsigned long long pk = lbest[wv][8*half + r];
                float bd2 = __uint_as_float((unsigned)(pk >> 32));
                bd2A[r] = bd2;
                tbdA[r] = 2.0f * sqrtf(bd2);
            }
        }

        if ((i & 15) == 15) {                                    // cross-strip best sharing
            if (lane < 16) {
                unsigned long long lb  = lbest[wv][m];
                unsigned long long old = atomicMin(&gbest[(size_t)b * NPTS + n0 + m], lb);
                if (old < lb) atomicMin(&lbest[wv][m], old);
            }
            asm volatile("s_wait_dscnt 0x0" ::: "memory");
#pragma unroll
            for (int r = 0; r < 8; ++r) {
                unsigned long long pk = lbest[wv][8*half + r];
                float bd2 = __uint_as_float((unsigned)(pk >> 32));
                bd2A[r] = bd2;
                tbdA[r] = 2.0f * sqrtf(bd2);
            }
        }
    }
    asm volatile("s_wait_dscnt 0x0" ::: "memory");
    if (lane < 16) atomicMin(&gbest[(size_t)b * NPTS + n0 + m], lbest[wv][m]);
}

// ---------------- kernel 3: finalize outputs for winning face ----------------
__global__ __launch_bounds__(256) void k_final(
    const float* __restrict__ coords, const float* __restrict__ smpl_V,
    const int* __restrict__ smpl_F, const float* __restrict__ can_V,
    const unsigned long long* __restrict__ gbest, float* __restrict__ out)
{
    int p = blockIdx.x * 256 + threadIdx.x;
    if (p >= NPOINTS) return;
    int b = p >> 10;
    unsigned fid = (unsigned)(gbest[p] & 0xFFFFFFFFull);
    float px = coords[p*3+0], py = coords[p*3+1], pz = coords[p*3+2];
    int ia = smpl_F[fid*3+0], ib = smpl_F[fid*3+1], ic = smpl_F[fid*3+2];
    const float* Vb = smpl_V + (size_t)b * V_NUM * 3;
    float ax = Vb[ia*3+0], ay = Vb[ia*3+1], az = Vb[ia*3+2];
    float bx = Vb[ib*3+0], by = Vb[ib*3+1], bz = Vb[ib*3+2];
    float cx = Vb[ic*3+0], cy = Vb[ic*3+1], cz = Vb[ic*3+2];
    float u, v, w;
    bary_closest(px, py, pz, ax, ay, az, bx, by, bz, cx, cy, cz, u, v, w);
    float hx = u*ax + v*bx + w*cx;
    float hy = u*ay + v*by + w*cy;
    float hz = u*az + v*bz + w*cz;
    // normal = (hit - p) / max(||hit - p||, 1e-6)
    float fx = hx - px, fy = hy - py, fz = hz - pz;
    float nrm = sqrtf(fx*fx + fy*fy + fz*fz);
    float inv = 1.0f / fmaxf(nrm, 1e-6f);
    // sdf = sign(dot(p - hit, cross(b-a, c-a))) * sqrt(max(||p-hit||^2, 1e-12))
    float dx = px - hx, dy = py - hy, dz = pz - hz;
    float dist = sqrtf(fmaxf(dx*dx + dy*dy + dz*dz, 1e-12f));
    float e0x = bx-ax, e0y = by-ay, e0z = bz-az;
    float e1x = cx-ax, e1y = cy-ay, e1z = cz-az;
    float fnx = e0y*e1z - e0z*e1y;
    float fny = e0z*e1x - e0x*e1z;
    float fnz = e0x*e1y - e0y*e1x;
    float s = dx*fnx + dy*fny + dz*fnz;
    float sgn = (s > 0.f ? 1.f : 0.f) - (s < 0.f ? 1.f : 0.f);
    // out_coord = u*can_V[ia] + v*can_V[ib] + w*can_V[ic]
    float ocx = u*can_V[ia*3+0] + v*can_V[ib*3+0] + w*can_V[ic*3+0];
    float ocy = u*can_V[ia*3+1] + v*can_V[ib*3+1] + w*can_V[ic*3+1];
    float ocz = u*can_V[ia*3+2] + v*can_V[ib*3+2] + w*can_V[ic*3+2];
    float* out_coord = out;               // [2,1024,3]
    float* sdf_o     = out + 6144;        // [2,1024]
    float* normal_o  = out + 8192;        // [2,1024,3]
    float* z_o       = out + 14336;       // [2,1024,1]
    out_coord[p*3+0] = ocx; out_coord[p*3+1] = ocy; out_coord[p*3+2] = ocz;
    sdf_o[p] = sgn * dist;
    normal_o[p*3+0] = fx*inv; normal_o[p*3+1] = fy*inv; normal_o[p*3+2] = fz*inv;
    z_o[p] = pz;
}

extern "C" void kernel_launch(void* const* d_in, const int* in_sizes, int n_in,
                              void* d_out, int out_size, void* d_ws, size_t ws_size,
                              hipStream_t stream) {
    const float* coords = (const float*)d_in[0];
    const float* smpl_V = (const float*)d_in[1];
    const int*   smpl_F = (const int*)d_in[2];
    const float* can_V  = (const float*)d_in[3];
    float* out = (float*)d_out;
    char*  ws  = (char*)d_ws;

    unsigned long long* gbest = (unsigned long long*)ws;
    float4* cdata = (float4*)(ws + WS_CD);

    k_prep<<<(NBATCH * F_NUM + 255) / 256, 256, 0, stream>>>(
        smpl_V, smpl_F, cdata, gbest);
    k_search<<<(NGROUP * NSTRIP) / WPB, WPB * 32, 0, stream>>>(
        coords, gbest, cdata);
    k_final<<<(NPOINTS + 255) / 256, 256, 0, stream>>>(
        coords, smpl_V, smpl_F, can_V, gbest, out);
}